// NM_LSTM_53257594470616
// MI455X (gfx1250) — compile-verified
//
#include <hip/hip_runtime.h>
#include <math.h>
#include <stdint.h>

// Sizes fixed by the reference
#define B_  256
#define IN_ 512
#define HS_ 512
#define G4  2048   // 4*HS
#define JT  16     // hebb column tile per block (tile = 512 x 16 floats = 32 KB LDS)

typedef __attribute__((ext_vector_type(2))) float v2f;
typedef __attribute__((ext_vector_type(8))) float v8f;
typedef int vi4_ __attribute__((vector_size(16)));   // payload type the async builtin wants

__device__ __forceinline__ float sigmoidf_(float x) {
    return 1.0f / (1.0f + __expf(-x));
}

// ---------------------------------------------------------------------------
// Kernel 1: gates = [x|h] @ [W;U] + bias, fp32 WMMA (16x16x4), one wave per
// 16x16 output tile. i/f/o columns get sigmoid applied; g columns stay raw.
// grid = (16, 16), block = 256 (8 waves); wave w handles n0 = 128*bx + 16*w.
// ---------------------------------------------------------------------------
__global__ __launch_bounds__(256) void nm_lstm_gates_wmma(
    const float* __restrict__ x, const float* __restrict__ h,
    const float* __restrict__ W, const float* __restrict__ U,
    const float* __restrict__ bias, float* __restrict__ gates_out)
{
    const int lane = threadIdx.x & 31;
    const int wave = threadIdx.x >> 5;
    const int m0 = blockIdx.y * 16;
    const int n0 = blockIdx.x * 128 + wave * 16;
    const int mi = lane & 15;      // A row / B col within tile
    const int kh = lane >> 4;      // K half (layout: VGPR0 = K0|K2, VGPR1 = K1|K3)
    const int row = m0 + mi;
    const int col = n0 + mi;

    v8f acc = {};

    // --- x @ W half (K = 0..511) ---
    {
        const float* Ar = x + (size_t)row * IN_;
        const float* Bp = W + col;
        for (int k = 0; k < IN_; k += 4) {
            v2f a, b;
            a.x = Ar[k + 2 * kh];
            a.y = Ar[k + 2 * kh + 1];
            b.x = Bp[(size_t)(k + 2 * kh) * G4];
            b.y = Bp[(size_t)(k + 2 * kh + 1) * G4];
            acc = __builtin_amdgcn_wmma_f32_16x16x4_f32(
                false, a, false, b, (short)0, acc, false, false);
        }
    }
    // --- h @ U half (K = 512..1023) ---
    {
        const float* Ar = h + (size_t)row * HS_;
        const float* Bp = U + col;
        for (int k = 0; k < HS_; k += 4) {
            v2f a, b;
            a.x = Ar[k + 2 * kh];
            a.y = Ar[k + 2 * kh + 1];
            b.x = Bp[(size_t)(k + 2 * kh) * G4];
            b.y = Bp[(size_t)(k + 2 * kh + 1) * G4];
            acc = __builtin_amdgcn_wmma_f32_16x16x4_f32(
                false, a, false, b, (short)0, acc, false, false);
        }
    }

    // Epilogue: bias, sigmoid for i/f/o regions (0,1,3), raw for g (2).
    const int region = n0 >> 9;          // whole 16-wide tile lies in one region
    const float bn = bias[col];
    #pragma unroll
    for (int r = 0; r < 8; ++r) {
        const int om = m0 + r + 8 * kh;  // C layout: lanes 16-31 hold M = r+8
        float v = acc[r] + bn;
        if (region != 2) v = sigmoidf_(v);
        gates_out[(size_t)om * G4 + col] = v;
    }
}

// ---------------------------------------------------------------------------
// Kernel 2: hebb streamer. One block per (batch b, JT-col tile).
//   phase A: async-DMA the 512 x JT hebb tile into LDS (ASYNCcnt path);
//            concurrently load h[b] -> LDS and block-reduce dot(h, nm_w).
//   phase B: g_sum[j] = sum_i h[i]*tile[i,j] from LDS;
//            g[j] = tanh(gates_raw_g + alpha[j]*g_sum); mg[j] = m_t[j]*g[j]
//   phase C: hebb_new[b,i,j] = clip(tile[i,j] + h[i]*mg[j], -2, 2) -> global
// HBM traffic: exactly one read + one write of hebb. grid = (32, 256).
// ---------------------------------------------------------------------------
__global__ __launch_bounds__(256) void nm_lstm_hebb(
    const float* __restrict__ h_t, const float* __restrict__ hebb,
    const float* __restrict__ alpha, const float* __restrict__ nm_w,
    const float* __restrict__ nm_b, const float* __restrict__ mt_w,
    const float* __restrict__ mt_b, const float* __restrict__ gates_raw,
    float* __restrict__ g_ws, float* __restrict__ hebb_out)
{
    const int b  = blockIdx.y;
    const int j0 = blockIdx.x * JT;
    const int t  = threadIdx.x;

    __shared__ __align__(16) float tile[HS_ * JT];   // 32 KB
    __shared__ float h_sh[HS_];
    __shared__ float red[8];
    __shared__ __align__(16) float part[64][JT];     // 4 KB
    __shared__ __align__(16) float mg_sh[JT];
    __shared__ float ms_sh;

    const float* gbase = hebb + (size_t)b * HS_ * HS_ + j0;  // row i at +i*HS_

    // --- Phase A: stage tile into LDS ------------------------------------
#if defined(__gfx1250__) && __has_builtin(__builtin_amdgcn_global_load_async_to_lds_b128)
    {
        // 2048 b128 transfers total; thread t issues 8 (stride 256 transfers).
        const int row0 = t >> 2;                  // 0..63
        const int seg  = t & 3;                   // 16B segment within a row
        const char* gp = (const char*)(gbase + (size_t)row0 * HS_) + seg * 16;
        unsigned lofs  = (unsigned)t * 16u;       // LDS byte offset of transfer t
        #pragma unroll
        for (int it = 0; it < 8; ++it) {
            __builtin_amdgcn_global_load_async_to_lds_b128(
                (__attribute__((address_space(1))) vi4_*)(uintptr_t)gp,
                (__attribute__((address_space(3))) vi4_*)(uintptr_t)
                    ((const char*)tile + lofs),
                0, 0);
            gp   += (size_t)64 * HS_ * 4;         // advance 64 rows
            lofs += 256u * 16u;
        }
    }
#define NM_ASYNC_STAGED 1
#else
    {
        float4* t4 = (float4*)tile;
        for (int idx = t; idx < (HS_ * JT) / 4; idx += 256) {
            const int row = idx >> 2, seg = idx & 3;
            t4[idx] = *(const float4*)(gbase + (size_t)row * HS_ + seg * 4);
        }
    }
#endif

    // h[b] -> LDS + partial dot(h, nm_w[0]) (overlaps with async DMA)
    float p = 0.0f;
    for (int i = t; i < HS_; i += 256) {
        float hv = h_t[(size_t)b * HS_ + i];
        h_sh[i] = hv;
        p += hv * nm_w[i];
    }
    #pragma unroll
    for (int off = 16; off > 0; off >>= 1) p += __shfl_down(p, off, 32);
    if ((t & 31) == 0) red[t >> 5] = p;

#ifdef NM_ASYNC_STAGED
    asm volatile("s_wait_asynccnt 0x0" ::: "memory");   // tile DMA complete
#endif
    __syncthreads();
    if (t == 0) {
        float s = 0.0f;
        #pragma unroll
        for (int w = 0; w < 8; ++w) s += red[w];
        ms_sh = tanhf(s + nm_b[0]);                     // NM == 1
    }
    __syncthreads();
    const float ms = ms_sh;

    // --- Phase B: column sums from LDS -----------------------------------
    const int tj = t & 3;     // float4 column group (4 cols)
    const int ti = t >> 2;    // row phase 0..63
    const float4* tile4 = (const float4*)tile;
    float4 acc = make_float4(0.f, 0.f, 0.f, 0.f);
    #pragma unroll
    for (int i = ti; i < HS_; i += 64) {
        float4 v = tile4[i * (JT / 4) + tj];
        float hv = h_sh[i];
        acc.x += hv * v.x; acc.y += hv * v.y;
        acc.z += hv * v.z; acc.w += hv * v.w;
    }
    ((float4*)part[ti])[tj] = acc;
    __syncthreads();

    if (t < JT) {
        float s = 0.0f;
        for (int w = 0; w < 64; ++w) s += part[w][t];
        const int j = j0 + t;
        float g = tanhf(gates_raw[(size_t)b * G4 + 2 * HS_ + j] + alpha[j] * s);
        g_ws[(size_t)b * HS_ + j] = g;
        float mt = ms * mt_w[j] + mt_b[j];
        mg_sh[t] = mt * g;
    }
    __syncthreads();

    // --- Phase C: rank-1 update + clip, stream to global ------------------
    const float4 mg = ((const float4*)mg_sh)[tj];
    float4* out4 = (float4*)(hebb_out + (size_t)b * HS_ * HS_ + j0);
    #pragma unroll
    for (int i = ti; i < HS_; i += 64) {
        float4 v = tile4[i * (JT / 4) + tj];
        float hv = h_sh[i];
        float4 r;
        r.x = fminf(2.0f, fmaxf(-2.0f, v.x + hv * mg.x));
        r.y = fminf(2.0f, fmaxf(-2.0f, v.y + hv * mg.y));
        r.z = fminf(2.0f, fmaxf(-2.0f, v.z + hv * mg.z));
        r.w = fminf(2.0f, fmaxf(-2.0f, v.w + hv * mg.w));
        out4[(size_t)i * (HS_ / 4) + tj] = r;
    }
}

// ---------------------------------------------------------------------------
// Kernel 3: LSTM cell pointwise. grid = 512 blocks x 256 threads.
// ---------------------------------------------------------------------------
__global__ __launch_bounds__(256) void nm_lstm_cell(
    const float* __restrict__ gates_act, const float* __restrict__ g_ws,
    const float* __restrict__ c_t,
    float* __restrict__ h_new, float* __restrict__ c_new)
{
    const int idx = blockIdx.x * 256 + threadIdx.x;   // 0 .. B*HS-1
    const int b = idx >> 9;
    const int j = idx & (HS_ - 1);
    const float* gb = gates_act + (size_t)b * G4;
    const float i_t = gb[j];
    const float f_t = gb[HS_ + j];
    const float o_t = gb[3 * HS_ + j];
    const float g_t = g_ws[idx];
    const float c   = f_t * c_t[idx] + i_t * g_t;
    c_new[idx] = c;
    h_new[idx] = o_t * tanhf(c);
}

// ---------------------------------------------------------------------------
// Launch
// inputs: 0:x_t 1:h_t 2:c_t 3:hebb 4:W 5:U 6:bias 7:alpha 8:nm_w 9:nm_b
//         10:mt_w 11:mt_b
// d_out : h_new (B*HS) | c_new (B*HS) | hebb_new (B*HS*HS)
// ---------------------------------------------------------------------------
extern "C" void kernel_launch(void* const* d_in, const int* in_sizes, int n_in,
                              void* d_out, int out_size, void* d_ws, size_t ws_size,
                              hipStream_t stream) {
    const float* x    = (const float*)d_in[0];
    const float* h    = (const float*)d_in[1];
    const float* c    = (const float*)d_in[2];
    const float* hebb = (const float*)d_in[3];
    const float* W    = (const float*)d_in[4];
    const float* U    = (const float*)d_in[5];
    const float* bias = (const float*)d_in[6];
    const float* alph = (const float*)d_in[7];
    const float* nm_w = (const float*)d_in[8];
    const float* nm_b = (const float*)d_in[9];
    const float* mt_w = (const float*)d_in[10];
    const float* mt_b = (const float*)d_in[11];

    float* out      = (float*)d_out;
    float* h_new    = out;
    float* c_new    = out + (size_t)B_ * HS_;
    float* hebb_new = out + (size_t)2 * B_ * HS_;

    float* gates_ws = (float*)d_ws;                       // B x 4HS
    float* g_ws     = gates_ws + (size_t)B_ * G4;         // B x HS

    dim3 gA(G4 / 128, B_ / 16);            // (16, 16)
    nm_lstm_gates_wmma<<<gA, 256, 0, stream>>>(x, h, W, U, bias, gates_ws);

    dim3 gB(HS_ / JT, B_);                 // (32, 256)
    nm_lstm_hebb<<<gB, 256, 0, stream>>>(h, hebb, alph, nm_w, nm_b, mt_w, mt_b,
                                         gates_ws, g_ws, hebb_new);

    nm_lstm_cell<<<(B_ * HS_) / 256, 256, 0, stream>>>(gates_ws, g_ws, c,
                                                       h_new, c_new);
}